// ImageRenderer_3358664425510
// MI455X (gfx1250) — compile-verified
//
#include <hip/hip_runtime.h>
#include <math.h>

#define BIG_Z 1000000.0f
#define EPS_D 1e-8f
#define CHF 128          // faces per LDS chunk: 128*64B = 8 KB
#define RS_BLOCK 256     // 8 wave32s per workgroup

typedef __attribute__((ext_vector_type(2))) float v2f;
typedef __attribute__((ext_vector_type(4))) float v4f;
typedef __attribute__((ext_vector_type(8))) float v8f;
typedef __attribute__((ext_vector_type(4))) unsigned v4u;
typedef __attribute__((ext_vector_type(8))) unsigned v8u;

// ---------------------------------------------------------------------------
// Kernel 1: per (batch, face) precompute.
// Record (16 floats): [a0,b0,c0,0, a1,b1,c1,0, a2,b2,c2,0, az,bz,cz,0]
// such that wK(px,py) = aK*px + bK*py + cK  and z(px,py) = az*px + bz*py + cz.
// UV record (8 floats): [u0,v0, u1,v1, u2,v2, 0,0] from the first camera.
// ---------------------------------------------------------------------------
__global__ void prep_faces_kernel(const float* __restrict__ cam,
                                  const float* __restrict__ verts,
                                  const float* __restrict__ cam_new,
                                  const int*   __restrict__ faces,
                                  float* __restrict__ coefs,
                                  float* __restrict__ uvs,
                                  int B, int N, int F)
{
    int gid = blockIdx.x * blockDim.x + threadIdx.x;
    if (gid >= B * F) return;
    int b = gid / F;
    int f = gid - b * F;

    float c0 = cam[b*3+0],     c1 = cam[b*3+1],     c2 = cam[b*3+2];
    float n0 = cam_new[b*3+0], n1 = cam_new[b*3+1], n2 = cam_new[b*3+2];

    float vx[3], vy[3], vz[3], uu[3], vv[3];
#pragma unroll
    for (int k = 0; k < 3; ++k) {
        int vi = faces[f*3+k];
        const float* V = verts + ((size_t)b * N + vi) * 3;
        float X = V[0], Y = V[1], Z = V[2];
        // transform under cam_new (raster geometry)
        float tx = n0 * (X + n1), ty = n0 * (Y + n2), tz = n0 * Z;
        vx[k] = fminf(fmaxf(tx, -1.f), 1.f);
        vy[k] = fminf(fmaxf(-ty, -1.f), 1.f);
        vz[k] = -tz + 10.f;
        // transform under cam (UV attributes)
        float sx = c0 * (X + c1), sy = c0 * (Y + c2);
        uu[k] = fminf(fmaxf(sx, -1.f), 1.f);
        vv[k] = fminf(fmaxf(-sy, -1.f), 1.f);
    }

    float e1x = vx[1]-vx[0], e1y = vy[1]-vy[0];
    float e2x = vx[2]-vx[0], e2y = vy[2]-vy[0];
    float d = e1x*e2y - e2x*e1y;
    bool valid = fabsf(d) > EPS_D;
    float inv = 1.f / (valid ? d : 1.f);

    float a1 =  e2y*inv, b1 = -e2x*inv, cc1 = (e2x*vy[0] - vx[0]*e2y)*inv;
    float a2 = -e1y*inv, b2 =  e1x*inv, cc2 = (vx[0]*e1y - e1x*vy[0])*inv;
    float a0 = -(a1+a2), b0 = -(b1+b2), cc0 = 1.f - cc1 - cc2;
    float az = a0*vz[0] + a1*vz[1] + a2*vz[2];
    float bz = b0*vz[0] + b1*vz[1] + b2*vz[2];
    float cz = cc0*vz[0] + cc1*vz[1] + cc2*vz[2];
    if (!valid) {   // never "inside"
        a0=b0=0.f; cc0=-1.f;
        a1=b1=0.f; cc1=-1.f;
        a2=b2=0.f; cc2=-1.f;
        az=bz=0.f; cz=BIG_Z;
    }

    float* rec = coefs + (size_t)gid * 16;
    v4f r0 = {a0, b0, cc0, 0.f};
    v4f r1 = {a1, b1, cc1, 0.f};
    v4f r2 = {a2, b2, cc2, 0.f};
    v4f r3 = {az, bz, cz,  0.f};
    *(v4f*)(rec + 0)  = r0;
    *(v4f*)(rec + 4)  = r1;
    *(v4f*)(rec + 8)  = r2;
    *(v4f*)(rec + 12) = r3;

    float* uvp = uvs + (size_t)gid * 8;
    v4f u0 = {uu[0], vv[0], uu[1], vv[1]};
    v4f u1 = {uu[2], vv[2], 0.f, 0.f};
    *(v4f*)(uvp + 0) = u0;
    *(v4f*)(uvp + 4) = u1;
}

// ---------------------------------------------------------------------------
// Bilinear fetch with zero padding outside the image.
// ---------------------------------------------------------------------------
__device__ __forceinline__ float fetch_px(const float* __restrict__ ic,
                                          int ix, int iy, int W, int H)
{
    bool val = (ix >= 0) && (ix < W) && (iy >= 0) && (iy < H);
    int ixc = min(max(ix, 0), W - 1);
    int iyc = min(max(iy, 0), H - 1);
    float v = ic[(size_t)iyc * W + ixc];
    return val ? v : 0.f;
}

// ---------------------------------------------------------------------------
// Kernel 2: WMMA rasterizer + fused grid_sample.
// Staging: each 8 KB face-record chunk is brought into LDS by the CDNA5 DMA
// paths — TDM (tensor_load_to_lds, TENSORcnt) moves the lower 4 KB with one
// descriptor issued by wave 0; global_load_async_to_lds_b128 (ASYNCcnt) moves
// the upper 4 KB (one b128 per thread). Chunks are double-buffered: the next
// chunk's DMA is in flight while the current one feeds the WMMA scan.
// ---------------------------------------------------------------------------
__global__ __launch_bounds__(RS_BLOCK) void raster_sample_kernel(
    const float* __restrict__ coefs,
    const float* __restrict__ uvs,
    const float* __restrict__ images,
    float* __restrict__ out,
    int B, int F, int H, int W)
{
    __shared__ __align__(16) float lds[2 * CHF * 16];   // 2 x 8 KB buffers

    const int P = H * W;
    const int pixPerBlock = (RS_BLOCK / 32) * 16;      // 128
    int blocksPerBatch = P / pixPerBlock;
    int b = blockIdx.x / blocksPerBatch;
    int pixBlock = blockIdx.x - b * blocksPerBatch;
    int lane = threadIdx.x & 31;
    int wave = threadIdx.x >> 5;
    int pixBase = pixBlock * pixPerBlock + wave * 16;

    // Build A matrix (16x4 f32): lanes 0-15 carry (px,py)=K0,K1 for M=0..15;
    // lanes 16-31 carry (1,0)=K2,K3.
    int mrow = lane & 15;
    int pA = pixBase + mrow;
    float pxA = ((float)(pA % W) + 0.5f) / (float)W * 2.f - 1.f;
    float pyA = ((float)(pA / W) + 0.5f) / (float)H * 2.f - 1.f;
    v2f A;
    A.x = (lane < 16) ? pxA : 1.f;
    A.y = (lane < 16) ? pyA : 0.f;

    const float* cb = coefs + (size_t)b * F * 16;

    // LDS byte address of the staging buffers (workgroup-relative).
    unsigned lds_base =
        (unsigned)(unsigned long long)(__attribute__((address_space(3))) float*)lds;

    // Stage one 8 KB chunk (base = first face) into buffer bufFloats.
    auto stage_chunk = [&](int base, int bufFloats) {
        unsigned long long ga =
            (unsigned long long)(unsigned long long)(cb + (size_t)base * 16);
        unsigned ldsOff = lds_base + (unsigned)bufFloats * 4u;
        if (threadIdx.x < 32) {
            // --- TDM descriptor: 1-row 2D tensor, 1024 f32 (lower 4 KB) ---
            v4u g0; v8u g1;
            g0[0] = 1u;                                   // count=1
            g0[1] = ldsOff;                               // lds_addr (bytes)
            g0[2] = (unsigned)ga;                         // global_addr[31:0]
            g0[3] = (unsigned)(ga >> 32) | 0x80000000u;   // addr[56:32] | type=2
            g1[0] = 2u << 16;                             // data_size = 4 bytes
            g1[1] = 1024u << 16;                          // tensor_dim0 = 1024
            g1[2] = 1u << 16;                             // tensor_dim1 = 1
            g1[3] = 1024u << 16;                          // tile_dim0 = 1024
            g1[4] = 1u;                                   // tile_dim1 = 1
            g1[5] = 1024u;                                // tensor_dim0_stride
            g1[6] = 0u;
            g1[7] = 0u;
            asm volatile("tensor_load_to_lds %0, %1"
                         :: "s"(g0), "s"(g1) : "memory");
        }
        // --- async global->LDS for the upper 4 KB: one b128 per thread ---
        unsigned ldsA = ldsOff + 4096u + ((unsigned)threadIdx.x << 4);
        unsigned vOff = (unsigned)threadIdx.x << 4;
        unsigned long long gaU = ga + 4096ull;
        asm volatile("global_load_async_to_lds_b128 %0, %1, %2"
                     :: "v"(ldsA), "v"(vOff), "s"(gaU) : "memory");
    };

    float bestZ[8];
    int   bestI[8];
#pragma unroll
    for (int v = 0; v < 8; ++v) { bestZ[v] = BIG_Z; bestI[v] = 0; }

    v8f Cz = {0.f, 0.f, 0.f, 0.f, 0.f, 0.f, 0.f, 0.f};

    stage_chunk(0, 0);   // prologue: chunk 0 -> buffer 0

    for (int base = 0; base < F; base += CHF) {
        int cur = (base / CHF) & 1;
        bool hasNext = (base + CHF) < F;
        if (hasNext) {
            stage_chunk(base + CHF, (cur ^ 1) * (CHF * 16));
            if (base + 2 * CHF < F)
                __builtin_prefetch(cb + (size_t)(base + 2 * CHF) * 16, 0, 0);
        }
        // In-order retirement: <=1 outstanding means the current chunk landed.
        if (hasNext) asm volatile("s_wait_asynccnt 0x1" ::: "memory");
        else         asm volatile("s_wait_asynccnt 0x0" ::: "memory");
        if (threadIdx.x < 32) {
            if (hasNext) __builtin_amdgcn_s_wait_tensorcnt(1);
            else         __builtin_amdgcn_s_wait_tensorcnt(0);
        }
        __syncthreads();

        const float* buf = lds + cur * (CHF * 16);
        for (int t = 0; t < CHF; t += 16) {
            const float* rec = buf + (size_t)(t + (lane & 15)) * 16;
            int off = (lane < 16) ? 0 : 2;
            v2f B0 = *(const v2f*)(rec + 0  + off);
            v2f B1 = *(const v2f*)(rec + 4  + off);
            v2f B2 = *(const v2f*)(rec + 8  + off);
            v2f B3 = *(const v2f*)(rec + 12 + off);

            v8f W0 = __builtin_amdgcn_wmma_f32_16x16x4_f32(false, A, false, B0, (short)0, Cz, false, false);
            v8f W1 = __builtin_amdgcn_wmma_f32_16x16x4_f32(false, A, false, B1, (short)0, Cz, false, false);
            v8f W2 = __builtin_amdgcn_wmma_f32_16x16x4_f32(false, A, false, B2, (short)0, Cz, false, false);
            v8f ZZ = __builtin_amdgcn_wmma_f32_16x16x4_f32(false, A, false, B3, (short)0, Cz, false, false);

            int fIdx = base + t + (lane & 15);
#pragma unroll
            for (int v = 0; v < 8; ++v) {
                bool inside = (W0[v] >= 0.f) && (W1[v] >= 0.f) && (W2[v] >= 0.f);
                float z = inside ? ZZ[v] : BIG_Z;
                if (z < bestZ[v]) { bestZ[v] = z; bestI[v] = fIdx; }
            }
        }
        __syncthreads();
    }

    // Min+index reduce across the 16 face-columns (16-lane halves of wave32).
#pragma unroll
    for (int v = 0; v < 8; ++v) {
        float z = bestZ[v]; int fi = bestI[v];
#pragma unroll
        for (int m = 1; m < 16; m <<= 1) {
            float oz = __shfl_xor(z, m, 32);
            int   oi = __shfl_xor(fi, m, 32);
            if (oz < z || (oz == z && oi < fi)) { z = oz; fi = oi; }
        }
        bestZ[v] = z; bestI[v] = fi;
    }

    // Finalize: lanes with (lane%16) < 8 each own one pixel.
    int s = lane & 15;
    if (s < 8) {
        float z = bestZ[0]; int fi = bestI[0];
#pragma unroll
        for (int v = 1; v < 8; ++v)
            if (s == v) { z = bestZ[v]; fi = bestI[v]; }

        int m = ((lane >> 4) << 3) + s;          // pixel index within wave's 16
        int p = pixBase + m;
        float px = ((float)(p % W) + 0.5f) / (float)W * 2.f - 1.f;
        float py = ((float)(p / W) + 0.5f) / (float)H * 2.f - 1.f;

        float u = 0.f, vq = 0.f;
        if (z < BIG_Z * 0.5f) {
            const float* rc = cb + (size_t)fi * 16;
            float w0 = rc[0]*px + rc[1]*py + rc[2];
            float w1 = rc[4]*px + rc[5]*py + rc[6];
            float w2 = rc[8]*px + rc[9]*py + rc[10];
            const float* uv = uvs + ((size_t)b * F + fi) * 8;
            u  = w0*uv[0] + w1*uv[2] + w2*uv[4];
            vq = w0*uv[1] + w1*uv[3] + w2*uv[5];
        }

        // grid_sample (align_corners=False, zero padding)
        float x = (u  + 1.f) * 0.5f * (float)W - 0.5f;
        float y = (vq + 1.f) * 0.5f * (float)H - 0.5f;
        float x0f = floorf(x), y0f = floorf(y);
        float wx1 = x - x0f, wx0 = 1.f - wx1;
        float wy1 = y - y0f, wy0 = 1.f - wy1;
        int ix0 = (int)x0f, iy0 = (int)y0f;

        const float* img = images + (size_t)b * 3 * P;
#pragma unroll
        for (int c = 0; c < 3; ++c) {
            const float* ic = img + (size_t)c * P;
            float f00 = fetch_px(ic, ix0,     iy0,     W, H);
            float f01 = fetch_px(ic, ix0 + 1, iy0,     W, H);
            float f10 = fetch_px(ic, ix0,     iy0 + 1, W, H);
            float f11 = fetch_px(ic, ix0 + 1, iy0 + 1, W, H);
            float r = wy0 * (wx0*f00 + wx1*f01) + wy1 * (wx0*f10 + wx1*f11);
            out[((size_t)(b*3 + c)) * P + p] = r;
        }
    }
}

// ---------------------------------------------------------------------------
extern "C" void kernel_launch(void* const* d_in, const int* in_sizes, int n_in,
                              void* d_out, int out_size, void* d_ws, size_t ws_size,
                              hipStream_t stream)
{
    (void)n_in; (void)out_size; (void)ws_size;
    const float* cam     = (const float*)d_in[0];
    const float* verts   = (const float*)d_in[1];
    const float* images  = (const float*)d_in[2];
    const float* cam_new = (const float*)d_in[3];
    const int*   faces   = (const int*)d_in[4];

    int B  = in_sizes[0] / 3;                 // 2
    int F  = in_sizes[4] / 3;                 // 10240
    int N  = in_sizes[1] / (3 * B);           // 5120
    int HW = in_sizes[2] / (3 * B);           // 65536
    int H  = (int)(sqrtf((float)HW) + 0.5f);  // 256
    int W  = H;

    float* coefs = (float*)d_ws;                     // B*F*16 floats
    float* uvs   = coefs + (size_t)B * F * 16;       // B*F*8 floats

    int total = B * F;
    prep_faces_kernel<<<(total + 255) / 256, 256, 0, stream>>>(
        cam, verts, cam_new, faces, coefs, uvs, B, N, F);

    int P = H * W;
    int blocks = B * (P / ((RS_BLOCK / 32) * 16));
    raster_sample_kernel<<<blocks, RS_BLOCK, 0, stream>>>(
        coefs, uvs, images, (float*)d_out, B, F, H, W);
}